// SpGAT_893353198192
// MI455X (gfx1250) — compile-verified
//
#include <hip/hip_runtime.h>
#include <cmath>

typedef float v2f __attribute__((ext_vector_type(2)));
typedef float v4f __attribute__((ext_vector_type(4)));
typedef float v8f __attribute__((ext_vector_type(8)));

#define N_NODES 20000
#define N_EDGES 320000
#define E_TOT   340000      // edges + self loops
#define IN_C    256
#define CAT     512         // 4 modules * 4 heads * 32
#define NH1     16          // fused head count layer 1
#define HIDC    32
#define OUT_C   40

__device__ __forceinline__ float lrelu(float v) { return v > 0.f ? v : 0.2f * v; }

__device__ __forceinline__ void edge_sd(int e, const int* __restrict__ ei, int& s, int& d) {
    if (e < N_EDGES) { s = ei[e]; d = ei[N_EDGES + e]; }
    else             { s = e - N_EDGES; d = s; }          // self loops appended
}

// float atomic max via orderable integer encoding (init with -inf)
__device__ __forceinline__ void atomicMaxF(float* addr, float v) {
    if (v >= 0.f) atomicMax((int*)addr, __float_as_int(v));
    else          atomicMin((unsigned int*)addr, (unsigned int)__float_as_int(v));
}

__global__ void fill_f32(float* __restrict__ p, float v, int n) {
    int i = blockIdx.x * blockDim.x + threadIdx.x;
    if (i < n) p[i] = v;
}

// ---------------- GEMM1: H[20000,512] = X[20000,256] @ Wcat[256,512] ----------------
// Wcat(k, col) = W1[m, k, j], m = col>>7, j = col&127  (module concat along columns)
__global__ __launch_bounds__(128) void gemm1_f32wmma(const float* __restrict__ X,
                                                     const float* __restrict__ W1,
                                                     float* __restrict__ H) {
    const int lane = threadIdx.x & 31;
    const int wave = threadIdx.x >> 5;
    const int row0 = blockIdx.x * 16;                      // 1250 tiles, exact
    const int col0 = (blockIdx.y * 4 + wave) * 16;         // 32 tiles, exact
    const int bk   = (lane >> 4) << 1;                     // lanes16-31 handle K+2,K+3
    const int ln   = lane & 15;
    const float* Xr = X + (row0 + ln) * IN_C;
    const float* Wc = W1 + ((col0 >> 7) << 15) + (col0 & 127) + ln;   // + k*128
    v8f acc = {};
    for (int k = 0; k < IN_C; k += 4) {
        v2f a, b;
        a[0] = Xr[k + bk];
        a[1] = Xr[k + bk + 1];
        b[0] = Wc[(k + bk) << 7];
        b[1] = Wc[(k + bk + 1) << 7];
        acc = __builtin_amdgcn_wmma_f32_16x16x4_f32(false, a, false, b,
                                                    (short)0, acc, false, false);
    }
    float* C = H + (row0 + ((lane >> 4) << 3)) * CAT + col0 + ln;
#pragma unroll
    for (int r = 0; r < 8; ++r) C[r * CAT] = acc[r];
}

// ---------------- GEMM2: G[20000,40] = H2[20000,512] @ W2[512,40] ----------------
__global__ __launch_bounds__(128) void gemm2_f32wmma(const float* __restrict__ H2,
                                                     const float* __restrict__ W2,
                                                     float* __restrict__ G) {
    const int gw = (blockIdx.x * 128 + threadIdx.x) >> 5;  // global wave id
    if (gw >= 1250 * 3) return;                            // wave-uniform exit
    const int lane = threadIdx.x & 31;
    const int mt = gw / 3, nt = gw - mt * 3;
    const int row0 = mt * 16, col0 = nt * 16;
    const int bk = (lane >> 4) << 1;
    const int ln = lane & 15;
    const float* Ar = H2 + (row0 + ln) * CAT;
    const int  cn = col0 + ln;
    const bool ok = cn < OUT_C;                            // pad N to 48, mask
    v8f acc = {};
    for (int k = 0; k < CAT; k += 4) {
        v2f a, b;
        a[0] = Ar[k + bk];
        a[1] = Ar[k + bk + 1];
        b[0] = ok ? W2[(k + bk) * OUT_C + cn] : 0.f;
        b[1] = ok ? W2[(k + bk + 1) * OUT_C + cn] : 0.f;
        acc = __builtin_amdgcn_wmma_f32_16x16x4_f32(false, a, false, b,
                                                    (short)0, acc, false, false);
    }
    if (ok) {
        float* C = G + (row0 + ((lane >> 4) << 3)) * OUT_C + cn;
#pragma unroll
        for (int r = 0; r < 8; ++r) C[r * OUT_C] = acc[r];
    }
}

// ---------------- Layer-1 attention ----------------
__global__ void att1_prep(const float* __restrict__ H, const float* __restrict__ as1,
                          const float* __restrict__ ad1,
                          float* __restrict__ es, float* __restrict__ ed) {
    int t = blockIdx.x * blockDim.x + threadIdx.x;
    if (t >= N_NODES * NH1) return;
    int n = t >> 4, h = t & 15;
    const v4f* hp = (const v4f*)(H + n * CAT + h * HIDC);
    const v4f* ap = (const v4f*)(as1 + h * HIDC);
    const v4f* bp = (const v4f*)(ad1 + h * HIDC);
    float s = 0.f, d = 0.f;
#pragma unroll
    for (int g = 0; g < HIDC / 4; ++g) {
        v4f v = hp[g], a = ap[g], b = bp[g];
#pragma unroll
        for (int j = 0; j < 4; ++j) { s += v[j] * a[j]; d += v[j] * b[j]; }
    }
    es[t] = s; ed[t] = d;
}

// one thread per edge: contiguous 4x b128 gathers of es/ed, 16 atomics
__global__ void att1_max(const int* __restrict__ ei, const float* __restrict__ es,
                         const float* __restrict__ ed, float* __restrict__ mx) {
    int e = blockIdx.x * blockDim.x + threadIdx.x;
    if (e >= E_TOT) return;
    int s, d; edge_sd(e, ei, s, d);
    const v4f* esp = (const v4f*)(es + s * NH1);
    const v4f* edp = (const v4f*)(ed + d * NH1);
    float* mp = mx + d * NH1;
#pragma unroll
    for (int g = 0; g < 4; ++g) {
        v4f a = esp[g], b = edp[g];
#pragma unroll
        for (int j = 0; j < 4; ++j) atomicMaxF(&mp[g * 4 + j], lrelu(a[j] + b[j]));
    }
}

__global__ void att1_sum(const int* __restrict__ ei, const float* __restrict__ es,
                         const float* __restrict__ ed, const float* __restrict__ mx,
                         float* __restrict__ sm) {
    int e = blockIdx.x * blockDim.x + threadIdx.x;
    if (e >= E_TOT) return;
    int s, d; edge_sd(e, ei, s, d);
    const v4f* esp = (const v4f*)(es + s * NH1);
    const v4f* edp = (const v4f*)(ed + d * NH1);
    const v4f* mp  = (const v4f*)(mx + d * NH1);
    float* sp = sm + d * NH1;
#pragma unroll
    for (int g = 0; g < 4; ++g) {
        v4f a = esp[g], b = edp[g], m = mp[g];
#pragma unroll
        for (int j = 0; j < 4; ++j)
            atomicAdd(&sp[g * 4 + j], expf(lrelu(a[j] + b[j]) - m[j]));
    }
}

// one thread per (edge, 4-channel group): 1 alpha, 1 b128 gather, 4 atomic adds
__global__ void att1_agg(const int* __restrict__ ei, const float* __restrict__ es,
                         const float* __restrict__ ed, const float* __restrict__ mx,
                         const float* __restrict__ sm, const float* __restrict__ H,
                         float* __restrict__ O) {
    int t = blockIdx.x * blockDim.x + threadIdx.x;
    if (t >= E_TOT * (CAT / 4)) return;                 // 43,520,000 threads
    int e = t >> 7, q = t & 127;
    int col = q << 2, h = q >> 3;
    int s, d; edge_sd(e, ei, s, d);
    int dh = d * NH1 + h;
    float v = lrelu(es[s * NH1 + h] + ed[dh]);
    float alpha = expf(v - mx[dh]) / (sm[dh] + 1e-16f);
    v4f hv = *(const v4f*)(H + s * CAT + col);
    float* op = O + d * CAT + col;
    atomicAdd(op + 0, alpha * hv[0]);
    atomicAdd(op + 1, alpha * hv[1]);
    atomicAdd(op + 2, alpha * hv[2]);
    atomicAdd(op + 3, alpha * hv[3]);
}

__global__ void elu_bias(float* __restrict__ O, const float* __restrict__ b1) {
    int t = blockIdx.x * blockDim.x + threadIdx.x;
    if (t >= N_NODES * CAT / 4) return;
    int i = t << 2;
    v4f v = *(const v4f*)(O + i);
    v4f b = *(const v4f*)(b1 + (i & 511));              // b1 flat == column order
#pragma unroll
    for (int j = 0; j < 4; ++j) {
        float u = v[j] + b[j];
        v[j] = u > 0.f ? u : expm1f(u);                 // ELU(alpha=1)
    }
    *(v4f*)(O + i) = v;
}

// ---------------- Layer-2 attention ----------------
__global__ void att2_prep(const float* __restrict__ G, const float* __restrict__ as2,
                          const float* __restrict__ ad2,
                          float* __restrict__ es, float* __restrict__ ed) {
    int n = blockIdx.x * blockDim.x + threadIdx.x;
    if (n >= N_NODES) return;
    const v4f* g = (const v4f*)(G + n * OUT_C);
    const v4f* a = (const v4f*)as2;
    const v4f* b = (const v4f*)ad2;
    float s = 0.f, d = 0.f;
#pragma unroll
    for (int q = 0; q < OUT_C / 4; ++q) {
        v4f v = g[q], aa = a[q], bb = b[q];
#pragma unroll
        for (int j = 0; j < 4; ++j) { s += v[j] * aa[j]; d += v[j] * bb[j]; }
    }
    es[n] = s; ed[n] = d;
}

__global__ void att2_max(const int* __restrict__ ei, const float* __restrict__ es,
                         const float* __restrict__ ed, float* __restrict__ mx) {
    int e = blockIdx.x * blockDim.x + threadIdx.x;
    if (e >= E_TOT) return;
    int s, d; edge_sd(e, ei, s, d);
    atomicMaxF(&mx[d], lrelu(es[s] + ed[d]));
}

__global__ void att2_sum(const int* __restrict__ ei, const float* __restrict__ es,
                         const float* __restrict__ ed, const float* __restrict__ mx,
                         float* __restrict__ sm) {
    int e = blockIdx.x * blockDim.x + threadIdx.x;
    if (e >= E_TOT) return;
    int s, d; edge_sd(e, ei, s, d);
    atomicAdd(&sm[d], expf(lrelu(es[s] + ed[d]) - mx[d]));
}

// one thread per (edge, 4-channel group): 40 cols = 10 groups (160B rows keep 16B align)
__global__ void att2_agg(const int* __restrict__ ei, const float* __restrict__ es,
                         const float* __restrict__ ed, const float* __restrict__ mx,
                         const float* __restrict__ sm, const float* __restrict__ G,
                         float* __restrict__ O) {
    int t = blockIdx.x * blockDim.x + threadIdx.x;
    if (t >= E_TOT * (OUT_C / 4)) return;               // 3,400,000 threads
    int e = t / 10, q = t - e * 10;
    int col = q << 2;
    int s, d; edge_sd(e, ei, s, d);
    float v = lrelu(es[s] + ed[d]);
    float alpha = expf(v - mx[d]) / (sm[d] + 1e-16f);
    v4f gv = *(const v4f*)(G + s * OUT_C + col);
    float* op = O + d * OUT_C + col;
    atomicAdd(op + 0, alpha * gv[0]);
    atomicAdd(op + 1, alpha * gv[1]);
    atomicAdd(op + 2, alpha * gv[2]);
    atomicAdd(op + 3, alpha * gv[3]);
}

__global__ void final_lsm(float* __restrict__ O, const float* __restrict__ b2) {
    int n = blockIdx.x * blockDim.x + threadIdx.x;
    if (n >= N_NODES) return;
    float r[OUT_C];
    float m = -3.4e38f;
#pragma unroll
    for (int c = 0; c < OUT_C; ++c) { r[c] = O[n * OUT_C + c] + b2[c]; m = fmaxf(m, r[c]); }
    float s = 0.f;
#pragma unroll
    for (int c = 0; c < OUT_C; ++c) s += expf(r[c] - m);
    float lse = m + logf(s);
#pragma unroll
    for (int c = 0; c < OUT_C; ++c) O[n * OUT_C + c] = r[c] - lse;
}

extern "C" void kernel_launch(void* const* d_in, const int* in_sizes, int n_in,
                              void* d_out, int out_size, void* d_ws, size_t ws_size,
                              hipStream_t stream) {
    (void)in_sizes; (void)n_in; (void)out_size; (void)ws_size;
    const float* x   = (const float*)d_in[0];
    const int*   ei  = (const int*)  d_in[1];
    const float* W1  = (const float*)d_in[2];
    const float* as1 = (const float*)d_in[3];
    const float* ad1 = (const float*)d_in[4];
    const float* b1  = (const float*)d_in[5];
    const float* W2  = (const float*)d_in[6];
    const float* as2 = (const float*)d_in[7];
    const float* ad2 = (const float*)d_in[8];
    const float* b2  = (const float*)d_in[9];
    float* out = (float*)d_out;
    float* ws  = (float*)d_ws;

    // workspace layout (floats): ~22.6M floats (~91 MB) — fits L2 together with inputs
    float* H   = ws;
    float* O1  = H   + (size_t)N_NODES * CAT;
    float* ES1 = O1  + (size_t)N_NODES * CAT;
    float* ED1 = ES1 + (size_t)N_NODES * NH1;
    float* MX1 = ED1 + (size_t)N_NODES * NH1;
    float* SM1 = MX1 + (size_t)N_NODES * NH1;
    float* G   = SM1 + (size_t)N_NODES * NH1;
    float* ES2 = G   + (size_t)N_NODES * OUT_C;
    float* ED2 = ES2 + N_NODES;
    float* MX2 = ED2 + N_NODES;
    float* SM2 = MX2 + N_NODES;

    auto nb = [](int n, int b) { return (n + b - 1) / b; };

    fill_f32<<<nb(N_NODES * NH1, 256), 256, 0, stream>>>(MX1, -INFINITY, N_NODES * NH1);
    fill_f32<<<nb(N_NODES * NH1, 256), 256, 0, stream>>>(SM1, 0.f, N_NODES * NH1);
    fill_f32<<<nb(N_NODES * CAT, 256), 256, 0, stream>>>(O1, 0.f, N_NODES * CAT);
    fill_f32<<<nb(N_NODES, 256), 256, 0, stream>>>(MX2, -INFINITY, N_NODES);
    fill_f32<<<nb(N_NODES, 256), 256, 0, stream>>>(SM2, 0.f, N_NODES);
    fill_f32<<<nb(N_NODES * OUT_C, 256), 256, 0, stream>>>(out, 0.f, N_NODES * OUT_C);

    gemm1_f32wmma<<<dim3(N_NODES / 16, (CAT / 16) / 4), 128, 0, stream>>>(x, W1, H);
    att1_prep<<<nb(N_NODES * NH1, 256), 256, 0, stream>>>(H, as1, ad1, ES1, ED1);
    att1_max<<<nb(E_TOT, 256), 256, 0, stream>>>(ei, ES1, ED1, MX1);
    att1_sum<<<nb(E_TOT, 256), 256, 0, stream>>>(ei, ES1, ED1, MX1, SM1);
    att1_agg<<<nb(E_TOT * (CAT / 4), 256), 256, 0, stream>>>(ei, ES1, ED1, MX1, SM1, H, O1);
    elu_bias<<<nb(N_NODES * CAT / 4, 256), 256, 0, stream>>>(O1, b1);

    gemm2_f32wmma<<<nb(1250 * 3 * 32, 128), 128, 0, stream>>>(O1, W2, G);
    att2_prep<<<nb(N_NODES, 256), 256, 0, stream>>>(G, as2, ad2, ES2, ED2);
    att2_max<<<nb(E_TOT, 256), 256, 0, stream>>>(ei, ES2, ED2, MX2);
    att2_sum<<<nb(E_TOT, 256), 256, 0, stream>>>(ei, ES2, ED2, MX2, SM2);
    att2_agg<<<nb(E_TOT * (OUT_C / 4), 256), 256, 0, stream>>>(ei, ES2, ED2, MX2, SM2, G, out);
    final_lsm<<<nb(N_NODES, 256), 256, 0, stream>>>(out, b2);
}